// SSIMLoss3D_37503654429361
// MI455X (gfx1250) — compile-verified
//
#include <hip/hip_runtime.h>
#include <hip/hip_bf16.h>

// SSIM 3D loss for MI455X (gfx1250).
// Separable 11x11 Gaussian conv done with V_WMMA_F32_16X16X4_F32 chains
// (full fp32 precision, needed because sigma = E[x^2]-mu^2 cancels against
// C2=9e-4). Each wave computes one 16x16 output tile; 5 channels
// (i1, i2, i1^2, i2^2, i1*i2) share weight fragments and LDS-staged inputs.

typedef float v2f __attribute__((ext_vector_type(2)));
typedef float v8f __attribute__((ext_vector_type(8)));

#define WSZ   11
#define SSIM_C1 1e-4f
#define SSIM_C2 9e-4f

__device__ __forceinline__ int reflect256(int p) {
    if (p < 0) p = -p;
    if (p > 255) p = 510 - p;
    return p;
}

__global__ __launch_bounds__(128)
void ssim_tile_kernel(const float* __restrict__ img1,
                      const float* __restrict__ img2,
                      const float* __restrict__ win,
                      float* __restrict__ partials)
{
    __shared__ float sg[16];
    __shared__ __align__(16) float in1[32][76];   // 26 used rows + zero pad, 74 used cols
    __shared__ __align__(16) float in2[32][76];
    __shared__ __align__(16) float hbuf[4][5][16][28]; // per-wave, transposed H: [x][row]
    __shared__ float wsum[4];

    const int tid  = threadIdx.x;
    const int wave = tid >> 5;
    const int lane = tid & 31;

    // 1D gaussian from the 2D window: since w2 = outer(g,g) and sum(g)=1,
    // g[i] = sum_j w2[i][j].
    if (tid < WSZ) {
        float s = 0.f;
        #pragma unroll
        for (int j = 0; j < WSZ; ++j) s += win[tid * WSZ + j];
        sg[tid] = s;
    }

    // tile coordinates: 64 workgroups per frame (4 in x covering 64 px, 16 in y)
    const int wgid  = blockIdx.x;
    const int frame = wgid >> 6;
    const int rem   = wgid & 63;
    const int wx    = rem & 3;
    const int wy    = rem >> 2;
    const int x0    = wx * 64;
    const int y0    = wy * 16;
    const size_t fbase = (size_t)frame * 65536;

    // Cooperative load with reflect padding + nan_to_num. Rows >=26 and
    // cols >=74 are zero (they feed only zero-weight K-padding of the WMMAs).
    for (int idx = tid; idx < 32 * 76; idx += 128) {
        int r = idx / 76;
        int c = idx - r * 76;
        float v1 = 0.f, v2 = 0.f;
        if (r < 26 && c < 74) {
            int gy = reflect256(y0 - 5 + r);
            int gx = reflect256(x0 - 5 + c);
            size_t o = fbase + (size_t)gy * 256 + (size_t)gx;
            v1 = img1[o];
            v2 = img2[o];
            v1 = (v1 == v1) ? v1 : 0.f;   // nan_to_num
            v2 = (v2 == v2) ? v2 : 0.f;
        }
        in1[r][c] = v1;
        in2[r][c] = v2;
    }
    __syncthreads();

    // Per-lane banded-Gaussian fragments. Identical layout serves as the
    // B operand of the horizontal pass and the A operand of the vertical pass:
    //   value = g[(4j + klo + {0,1}) - (lane & 15)]  (0 outside [0,11))
    const int m   = lane & 15;
    const int klo = (lane < 16) ? 0 : 2;
    v2f wf[7];
    #pragma unroll
    for (int j = 0; j < 7; ++j) {
        int q0 = 4 * j + klo - m;
        int q1 = q0 + 1;
        float g0 = (q0 >= 0 && q0 < WSZ) ? sg[q0] : 0.f;
        float g1 = (q1 >= 0 && q1 < WSZ) ? sg[q1] : 0.f;
        wf[j].x = g0;
        wf[j].y = g1;
    }

    const int xw = wave * 16;   // this wave's column window into the LDS tile

    // ---------------- Horizontal pass: H = In(16x28) x Wh(28x16) ------------
    // Two row batches cover the 26 halo rows (rows 26..31 are zero).
    #pragma unroll
    for (int b = 0; b < 2; ++b) {
        v8f hacc[5];
        #pragma unroll
        for (int ch = 0; ch < 5; ++ch) hacc[ch] = (v8f){};

        const int row = m + 16 * b;   // A-operand row for this lane
        #pragma unroll
        for (int j = 0; j < 7; ++j) {
            const int cb = xw + 4 * j + klo;
            v2f a1 = *(const v2f*)(&in1[row][cb]);
            v2f a2 = *(const v2f*)(&in2[row][cb]);
            v2f p11 = a1 * a1;
            v2f p22 = a2 * a2;
            v2f p12 = a1 * a2;
            hacc[0] = __builtin_amdgcn_wmma_f32_16x16x4_f32(false, a1,  false, wf[j], (short)0, hacc[0], false, false);
            hacc[1] = __builtin_amdgcn_wmma_f32_16x16x4_f32(false, a2,  false, wf[j], (short)0, hacc[1], false, false);
            hacc[2] = __builtin_amdgcn_wmma_f32_16x16x4_f32(false, p11, false, wf[j], (short)0, hacc[2], false, false);
            hacc[3] = __builtin_amdgcn_wmma_f32_16x16x4_f32(false, p22, false, wf[j], (short)0, hacc[3], false, false);
            hacc[4] = __builtin_amdgcn_wmma_f32_16x16x4_f32(false, p12, false, wf[j], (short)0, hacc[4], false, false);
        }

        // Store D (C-layout: VGPR i -> row 16b + i + (lane<16?0:8), col = m)
        // transposed into hbuf[x][row] so the vertical B-fragments are
        // contiguous ds_load_b64's. hbuf has 28 rows; rows 24..27 come from
        // batch 1 upper lanes (26,27 are exact zeros from the zero halo).
        const int rbase = 16 * b + ((lane < 16) ? 0 : 8);
        #pragma unroll
        for (int ch = 0; ch < 5; ++ch) {
            float* dst = &hbuf[wave][ch][m][rbase];
            v8f d = hacc[ch];
            dst[0] = d[0]; dst[1] = d[1]; dst[2] = d[2]; dst[3] = d[3];
            if (rbase + 7 < 28) {
                dst[4] = d[4]; dst[5] = d[5]; dst[6] = d[6]; dst[7] = d[7];
            }
        }
    }

    // ---------------- Vertical pass: V = Wv(16x28) x H(28x16) --------------
    v8f vacc[5];
    #pragma unroll
    for (int ch = 0; ch < 5; ++ch) {
        v8f acc = (v8f){};
        #pragma unroll
        for (int j = 0; j < 7; ++j) {
            v2f hv = *(const v2f*)(&hbuf[wave][ch][m][4 * j + klo]);
            acc = __builtin_amdgcn_wmma_f32_16x16x4_f32(false, wf[j], false, hv, (short)0, acc, false, false);
        }
        vacc[ch] = acc;
    }

    // ---------------- Pointwise SSIM + reduction ----------------------------
    float ssum = 0.f;
    #pragma unroll
    for (int i = 0; i < 8; ++i) {
        float mu1 = vacc[0][i], mu2 = vacc[1][i];
        float e11 = vacc[2][i], e22 = vacc[3][i], e12 = vacc[4][i];
        float mu1s = mu1 * mu1;
        float mu2s = mu2 * mu2;
        float mu12 = mu1 * mu2;
        float s1  = e11 - mu1s;
        float s2  = e22 - mu2s;
        float s12 = e12 - mu12;
        float num = (2.f * mu12 + SSIM_C1) * (2.f * s12 + SSIM_C2);
        float den = (mu1s + mu2s + SSIM_C1) * (s1 + s2 + SSIM_C2);
        ssum += num / den;
    }
    // wave32 reduce
    #pragma unroll
    for (int off = 16; off > 0; off >>= 1)
        ssum += __shfl_xor(ssum, off, 32);
    if (lane == 0) wsum[wave] = ssum;
    __syncthreads();
    if (tid == 0)
        partials[blockIdx.x] = wsum[0] + wsum[1] + wsum[2] + wsum[3];
}

__global__ __launch_bounds__(256)
void ssim_reduce_kernel(const float* __restrict__ partials, int n,
                        float* __restrict__ out, float inv_total)
{
    __shared__ float red[256];
    float s = 0.f;
    for (int i = threadIdx.x; i < n; i += 256) s += partials[i];
    red[threadIdx.x] = s;
    __syncthreads();
    for (int step = 128; step > 0; step >>= 1) {
        if (threadIdx.x < step) red[threadIdx.x] += red[threadIdx.x + step];
        __syncthreads();
    }
    if (threadIdx.x == 0) {
        float ssim = red[0] * inv_total;
        float loss = 1.f - ssim;
        loss = fminf(fmaxf(loss, 0.f), 1.f);
        out[0] = loss;
    }
}

extern "C" void kernel_launch(void* const* d_in, const int* in_sizes, int n_in,
                              void* d_out, int out_size, void* d_ws, size_t ws_size,
                              hipStream_t stream) {
    (void)n_in; (void)out_size; (void)ws_size;
    const float* img1 = (const float*)d_in[0];
    const float* img2 = (const float*)d_in[1];
    const float* win  = (const float*)d_in[2];

    const int total  = in_sizes[0];            // b*c*t*h*w = 16,777,216
    const int frames = total / (256 * 256);    // 256 frames of 256x256
    const int nblocks = frames * 64;           // 4 x-wgs * 16 y-wgs per frame

    float* partials = (float*)d_ws;            // nblocks floats of scratch

    ssim_tile_kernel<<<nblocks, 128, 0, stream>>>(img1, img2, win, partials);
    ssim_reduce_kernel<<<1, 256, 0, stream>>>(partials, nblocks, (float*)d_out,
                                              1.0f / (float)total);
}